// GCNetwork_39298950759069
// MI455X (gfx1250) — compile-verified
//
#include <hip/hip_runtime.h>
#include <hip/hip_bf16.h>

// ---------------------------------------------------------------------------
// GCN on MI455X (gfx1250, wave32).
// GEMM uses V_WMMA_F32_16X16X4_F32 (fp32 matches reference precision; GEMM is
// <5% of runtime anyway -- the edge scatter is the L2-bandwidth-bound phase).
// ---------------------------------------------------------------------------

typedef float v2f __attribute__((ext_vector_type(2)));
typedef float v8f __attribute__((ext_vector_type(8)));

#define FEAT 128
#define NGRAPH 512
#define NOUT 10

// ---------------- small utility kernels ----------------

__global__ void k_set(float* __restrict__ p, float val, int n) {
  int i = blockIdx.x * blockDim.x + threadIdx.x;
  if (i < n) p[i] = val;
}

__global__ void k_deg_accum(const int* __restrict__ dst, float* __restrict__ deg, int E) {
  int e = blockIdx.x * blockDim.x + threadIdx.x;
  if (e < E) atomicAdd(&deg[dst[e]], 1.0f);
}

__global__ void k_rsqrt(float* __restrict__ d, int n) {
  int i = blockIdx.x * blockDim.x + threadIdx.x;
  if (i < n) d[i] = rsqrtf(d[i]);
}

// ---------------- fp32 WMMA GEMM:  O[N,128] = H[N,128] @ W[128,128] ---------
// grid.x = N/16 row-strips, 256 threads = 8 waves; wave w owns 16 output cols.
// 32 x v_wmma_f32_16x16x4_f32 per wave (K = 128 in steps of 4).

__global__ __launch_bounds__(256) void k_gemm_wmma(const float* __restrict__ H,
                                                   const float* __restrict__ W,
                                                   float* __restrict__ O) {
  const int wave = threadIdx.x >> 5;      // 0..7 -> output column tile
  const int lane = threadIdx.x & 31;
  const int r    = lane & 15;             // row-in-tile / col-in-tile
  const int hi   = lane >> 4;             // half-wave selector
  const int m0   = blockIdx.x * 16;
  const int n0   = wave * 16;

  const float* __restrict__ hrow = H + (size_t)(m0 + r) * FEAT;

  v8f acc = {};
#pragma unroll
  for (int k0 = 0; k0 < FEAT; k0 += 4) {
    // A frag (16x4): lane l, vgpr j -> A[l%16][k0 + 2*(l/16) + j]  (contig pair)
    v2f a;
    a.x = hrow[k0 + 2 * hi + 0];
    a.y = hrow[k0 + 2 * hi + 1];
    // B frag (4x16): lane l, vgpr j -> B[k0 + 2*(l/16) + j][n0 + l%16]
    v2f b;
    b.x = W[(size_t)(k0 + 2 * hi + 0) * FEAT + n0 + r];
    b.y = W[(size_t)(k0 + 2 * hi + 1) * FEAT + n0 + r];
    // 8 args: (neg_a, A, neg_b, B, c_mod, C, reuse_a, reuse_b)
    acc = __builtin_amdgcn_wmma_f32_16x16x4_f32(false, a, false, b,
                                                (short)0, acc, false, false);
  }

  // D layout: vgpr v -> row m0 + v + 8*hi, col n0 + r
#pragma unroll
  for (int v = 0; v < 8; ++v) {
    O[(size_t)(m0 + v + 8 * hi) * FEAT + n0 + r] = acc[v];
  }
}

// ---------------- aggregation ----------------
// init: out[i][f] = hw[i][f] * dinv[i]^2 + bias[f]   (self-loop + bias, no atomics)
__global__ void k_agg_init(const float* __restrict__ hw, const float* __restrict__ dinv,
                           const float* __restrict__ bias, float* __restrict__ out, int n) {
  int t = blockIdx.x * blockDim.x + threadIdx.x;
  if (t >= n) return;
  int node = t >> 7;
  int f    = t & (FEAT - 1);
  float di = dinv[node];
  out[t] = hw[t] * di * di + bias[f];
}

// scatter: 32 lanes per edge, 4 coalesced float atomics per lane (128 feats)
__global__ __launch_bounds__(256) void k_agg_scatter(const float* __restrict__ hw,
                                                     const int* __restrict__ src,
                                                     const int* __restrict__ dst,
                                                     const float* __restrict__ dinv,
                                                     float* __restrict__ out, int E) {
  int t = blockIdx.x * blockDim.x + threadIdx.x;
  int e = t >> 5;
  if (e >= E) return;
  int lane = t & 31;
  int s = src[e];
  int d = dst[e];
  float norm = dinv[s] * dinv[d];
  const float* __restrict__ hs = hw + (size_t)s * FEAT;
  float* __restrict__ od = out + (size_t)d * FEAT;
#pragma unroll
  for (int j = 0; j < 4; ++j) {
    int f = lane + 32 * j;
    atomicAdd(&od[f], hs[f] * norm);
  }
}

// ---------------- pooling + output head ----------------

__global__ void k_pool(const float* __restrict__ h, const int* __restrict__ batch,
                       float* __restrict__ pooled, float* __restrict__ counts, int n) {
  int t = blockIdx.x * blockDim.x + threadIdx.x;
  if (t >= n) return;
  int node = t >> 7;
  int f    = t & (FEAT - 1);
  int g = batch[node];
  atomicAdd(&pooled[(size_t)g * FEAT + f], h[t]);
  if (f == 0) atomicAdd(&counts[g], 1.0f);
}

__global__ void k_final(const float* __restrict__ pooled, const float* __restrict__ counts,
                        const float* __restrict__ Wout, const float* __restrict__ bout,
                        float* __restrict__ out) {
  int t = blockIdx.x * blockDim.x + threadIdx.x;
  if (t >= NGRAPH * NOUT) return;
  int g = t / NOUT;
  int o = t % NOUT;
  float c = counts[g];
  c = c > 1.0f ? c : 1.0f;
  float acc = 0.0f;
#pragma unroll 8
  for (int k = 0; k < FEAT; ++k) acc += pooled[(size_t)g * FEAT + k] * Wout[k * NOUT + o];
  out[t] = acc / c + bout[o];
}

// ---------------- host-side orchestration ----------------

extern "C" void kernel_launch(void* const* d_in, const int* in_sizes, int n_in,
                              void* d_out, int out_size, void* d_ws, size_t ws_size,
                              hipStream_t stream) {
  const float* x     = (const float*)d_in[0];
  const int*   ei    = (const int*)d_in[1];   // [2, E] flat: row0=src, row1=dst
  const int*   batch = (const int*)d_in[2];
  const float* W_in  = (const float*)d_in[3];
  const float* b_in  = (const float*)d_in[4];
  const float* W_hid = (const float*)d_in[5]; // [3,128,128]
  const float* b_hid = (const float*)d_in[6]; // [3,128]
  const float* W_out = (const float*)d_in[7];
  const float* b_out = (const float*)d_in[8];
  float* out = (float*)d_out;

  const int N = in_sizes[0] / FEAT;           // 100000
  const int E = in_sizes[1] / 2;              // 1600000
  const int* src = ei;
  const int* dst = ei + E;

  // workspace carve-out (256B aligned): dinv, counts, pooled, bufA, bufB
  char* ws = (char*)d_ws;
  size_t off = 0;
  auto carve = [&](size_t bytes) -> char* {
    char* p = ws + off;
    off = (off + bytes + 255) & ~(size_t)255;
    return p;
  };
  float* dinv   = (float*)carve((size_t)N * 4);
  float* counts = (float*)carve((size_t)NGRAPH * 4);
  float* pooled = (float*)carve((size_t)NGRAPH * FEAT * 4);
  float* bufA   = (float*)carve((size_t)N * FEAT * 4);   // GEMM output
  float* bufB   = (float*)carve((size_t)N * FEAT * 4);   // aggregation output
  (void)ws_size;

  const int TB = 256;
  const int nNF = N * FEAT;

  // 1) degree (self-loop gives deg >= 1) -> dinv = rsqrt(deg)
  k_set<<<(N + TB - 1) / TB, TB, 0, stream>>>(dinv, 1.0f, N);
  k_deg_accum<<<(E + TB - 1) / TB, TB, 0, stream>>>(dst, dinv, E);
  k_rsqrt<<<(N + TB - 1) / TB, TB, 0, stream>>>(dinv, N);

  // 2) four GCN conv layers; ping-pong bufA (gemm out) / bufB (agg out)
  const float* cur = x;
  for (int layer = 0; layer < 4; ++layer) {
    const float* W = (layer == 0) ? W_in : (W_hid + (size_t)(layer - 1) * FEAT * FEAT);
    const float* b = (layer == 0) ? b_in : (b_hid + (size_t)(layer - 1) * FEAT);

    k_gemm_wmma<<<N / 16, 256, 0, stream>>>(cur, W, bufA);
    k_agg_init<<<(nNF + TB - 1) / TB, TB, 0, stream>>>(bufA, dinv, b, bufB, nNF);
    k_agg_scatter<<<((size_t)E * 32 + TB - 1) / TB, TB, 0, stream>>>(bufA, src, dst, dinv, bufB, E);
    cur = bufB;   // next layer reads bufB; its GEMM writes bufA (bufA dead by then)
  }

  // 3) global mean pool + output projection
  k_set<<<(NGRAPH * FEAT + TB - 1) / TB, TB, 0, stream>>>(pooled, 0.0f, NGRAPH * FEAT);
  k_set<<<(NGRAPH + TB - 1) / TB, TB, 0, stream>>>(counts, 0.0f, NGRAPH);
  k_pool<<<(nNF + TB - 1) / TB, TB, 0, stream>>>(bufB, batch, pooled, counts, nNF);
  k_final<<<(NGRAPH * NOUT + TB - 1) / TB, TB, 0, stream>>>(pooled, counts, W_out, b_out, out);
}